// KMerTransformer_28157805593024
// MI455X (gfx1250) — compile-verified
//
#include <hip/hip_runtime.h>

typedef __attribute__((ext_vector_type(2))) float v2f;
typedef __attribute__((ext_vector_type(8))) float v8f;

#define LOUT     99999     // output positions per batch (100001 - 3 + 1)
#define LQ       33333     // LOUT / 3 (rearrange block length)
#define NTILES   6250      // ceil(LOUT / 16)
#define XSTRIDE  400004    // floats per batch in x: 100001 * 4
#define NBATCH   8

__global__ __launch_bounds__(256)
void kmer_wmma_kernel(const float* __restrict__ x,
                      const float* __restrict__ w,
                      float* __restrict__ out)
{
    const int lane   = threadIdx.x & 31;
    const int wave   = threadIdx.x >> 5;
    const int wid    = blockIdx.x * (blockDim.x >> 5) + wave;   // 0 .. 49999
    const int b      = wid / NTILES;
    const int tile   = wid - b * NTILES;
    const int i0     = tile * 16;

    const int  lane16 = lane & 15;
    const bool hi     = lane >= 16;

    // ---------------- B fragments from W (logical [64 x 12]) ----------------
    // f32 4x16 B layout: N = lane&15; lanes 0-15 hold K={0,1} in v0,v1;
    // lanes 16-31 hold K={2,3}.  Each lane: one 8-byte load per (nt, kchunk).
    v2f bfrag[4][3];
#pragma unroll
    for (int nt = 0; nt < 4; ++nt) {
        const int n = nt * 16 + lane16;
#pragma unroll
        for (int t = 0; t < 3; ++t) {
            const int k = 4 * t + (hi ? 2 : 0);
            bfrag[nt][t] = *(const v2f*)(w + n * 12 + k);
        }
    }

    // ---------------- A fragments: im2col row i is x[b, i*4 .. i*4+11] ------
    // f32 16x4 A layout: M = lane&15; lanes 0-15 hold K={0,1}, lanes 16-31 K={2,3}.
    int irow = i0 + lane16;
    irow = (irow < LOUT) ? irow : (LOUT - 1);     // clamp last partial tile (VALU select)
    const float* xrow = x + (size_t)b * XSTRIDE + (size_t)irow * 4 + (hi ? 2 : 0);

    v2f afrag[3];
#pragma unroll
    for (int t = 0; t < 3; ++t)
        afrag[t] = *(const v2f*)(xrow + 4 * t);

    // Prefetch the next tile's portion of the x stream (global_prefetch_b8).
    __builtin_prefetch((const void*)(xrow + 64), 0, 1);

    // ---------------- 12 x V_WMMA_F32_16X16X4_F32 ---------------------------
    v8f acc[4];
#pragma unroll
    for (int nt = 0; nt < 4; ++nt) {
        v8f c = {};
#pragma unroll
        for (int t = 0; t < 3; ++t) {
            c = __builtin_amdgcn_wmma_f32_16x16x4_f32(
                    /*neg_a=*/false, afrag[t],
                    /*neg_b=*/false, bfrag[nt][t],
                    /*c_mod=*/(short)0, c,
                    /*reuse_a=*/false, /*reuse_b=*/false);
        }
        acc[nt] = c;
    }

    // ---------------- threshold-ReLU + rearrange scatter --------------------
    // D layout: vgpr r, lane<16 -> (M=r, N=lane); lane>=16 -> (M=r+8, N=lane-16).
    // Output [8,1,LOUT,64]: row p = (i%3)*LQ + i/3, channel c = nt*16 + N.
    float* outb = out + (size_t)b * (size_t)LOUT * 64;
#pragma unroll
    for (int r = 0; r < 8; ++r) {
        const int M = r + (hi ? 8 : 0);
        const int i = i0 + M;
        if (i < LOUT) {
            const int q  = i / 3;
            const int rm = i - q * 3;
            const int p  = rm * LQ + q;
            float* op = outb + (size_t)p * 64 + lane16;
#pragma unroll
            for (int nt = 0; nt < 4; ++nt) {
                float v = acc[nt][r] - 2.0f;
                v = (v > 0.0f) ? v : 0.0f;
                op[nt * 16] = v;
            }
        }
    }
}

extern "C" void kernel_launch(void* const* d_in, const int* in_sizes, int n_in,
                              void* d_out, int out_size, void* d_ws, size_t ws_size,
                              hipStream_t stream) {
    const float* x = (const float*)d_in[0];   // [8,1,100001,4] f32
    const float* w = (const float*)d_in[1];   // [64,1,3,4]     f32 (one-hot kmers)
    float* out = (float*)d_out;               // [8,1,99999,64] f32

    // 8 batches * 6250 tiles = 50,000 waves; 8 waves (256 threads) per block.
    dim3 grid(NBATCH * NTILES / 8), block(256);
    hipLaunchKernelGGL(kmer_wmma_kernel, grid, block, 0, stream, x, w, out);
}